// SplineConvNet_62414464745575
// MI455X (gfx1250) — compile-verified
//
#include <hip/hip_runtime.h>
#include <math.h>

#define N_NODES 10000
#define N_NODES_PAD 10016   // 313 m-pairs * 32 rows
#define N_EDGES 60000
#define KBINS   27
#define EPS_BN  1e-5f

typedef __attribute__((ext_vector_type(2))) float v2f;
typedef __attribute__((ext_vector_type(8))) float v8f;

// ---------------------------------------------------------------------------
// Per-edge degree-1 B-spline basis: 8 (weight, dst*32+kbin) pairs per edge.
// ---------------------------------------------------------------------------
__global__ void edge_basis_kernel(const float* __restrict__ eattr,
                                  const int* __restrict__ eidx,
                                  float* __restrict__ eb_w,
                                  int* __restrict__ eb_dk) {
  int e = blockIdx.x * blockDim.x + threadIdx.x;
  if (e >= N_EDGES) return;
  float fr[3];
  int i0[3];
#pragma unroll
  for (int d = 0; d < 3; ++d) {
    float p = eattr[e * 3 + d] * 2.0f;       // * (KDIM-1)
    float f = floorf(p);
    f = fminf(fmaxf(f, 0.0f), 1.0f);         // clip to [0, KDIM-2]
    i0[d] = (int)f;
    fr[d] = p - f;
  }
  int dst = eidx[N_EDGES + e];               // edge_index row 1
#pragma unroll
  for (int s = 0; s < 8; ++s) {
    int b0 = s & 1, b1 = (s >> 1) & 1, b2 = (s >> 2) & 1;
    float w = (b0 ? fr[0] : 1.0f - fr[0]) *
              (b1 ? fr[1] : 1.0f - fr[1]) *
              (b2 ? fr[2] : 1.0f - fr[2]);
    int k = (i0[0] + b0) + 3 * (i0[1] + b1) + 9 * (i0[2] + b2);
    eb_w[e * 8 + s]  = w;
    eb_dk[e * 8 + s] = (dst << 5) | k;       // kbin < 32
  }
}

// ---------------------------------------------------------------------------
// Scatter: A[node][kbin*cin + c] += w * x[src][c]   (one block per edge-slot)
// ---------------------------------------------------------------------------
__global__ void scatter_kernel(const float* __restrict__ x,
                               const float* __restrict__ eb_w,
                               const int* __restrict__ eb_dk,
                               const int* __restrict__ eidx,
                               float* __restrict__ Abuf,
                               int cin, int kcpad) {
  int slot = blockIdx.x;
  int e    = slot >> 3;
  float w  = eb_w[slot];
  int dk   = eb_dk[slot];
  int node = dk >> 5;
  int kb   = dk & 31;
  int src  = eidx[e];                        // edge_index row 0
  const float* xr = x + (size_t)src * cin;
  float* ar = Abuf + (size_t)node * kcpad + (size_t)kb * cin;
  for (int c = threadIdx.x; c < cin; c += blockDim.x) {
    __hip_atomic_fetch_add(ar + c, w * xr[c],
                           __ATOMIC_RELAXED, __HIP_MEMORY_SCOPE_AGENT);
  }
}

// ---------------------------------------------------------------------------
// Build transposed weight Wt[Npad][KPT]: rows n, cols k.
//   k < Kc            -> W[k][n]      (W flattened (27*cin, cout))
//   KcPad <= k, c<cin -> root[c][n]
//   else 0 (padding)
// ---------------------------------------------------------------------------
__global__ void build_wt_kernel(const float* __restrict__ W,
                                const float* __restrict__ root,
                                float* __restrict__ Wt,
                                int cout, int cin, int Kc, int KcPad,
                                int KPT, int total) {
  for (int idx = blockIdx.x * blockDim.x + threadIdx.x; idx < total;
       idx += gridDim.x * blockDim.x) {
    int n = idx / KPT;
    int k = idx - n * KPT;
    float v = 0.0f;
    if (n < cout) {
      if (k < Kc) {
        v = W[(size_t)k * cout + n];
      } else if (k >= KcPad) {
        int c = k - KcPad;
        if (c < cin) v = root[(size_t)c * cout + n];
      }
    }
    Wt[idx] = v;
  }
}

// ---------------------------------------------------------------------------
// WMMA f32 GEMM with 2x2 fragment blocking: each wave computes a 32x32
// output block (2 m-tiles x 2 n-tiles): 4 WMMAs per 4 b64 loads.
// Waves of a block share the m-pair, take consecutive n-pairs -> A frags
// hit the WGP cache for 3 of 4 waves. Grid parallelism is huge (~2.5k
// waves), so trade occupancy for VGPRs (waves_per_eu 1..2) to let the
// scheduler keep all fragment loads of a step (and the next) in flight.
// f32 16x16x4 fragment layout (ISA 7.12.2): lane<16 holds K=k0,k0+1;
// lane>=16 holds K=k0+2,k0+3; row/col = lane%16.
// ---------------------------------------------------------------------------
__global__ __attribute__((amdgpu_flat_work_group_size(128, 128)))
__attribute__((amdgpu_waves_per_eu(1, 2))) void spline_gemm_wmma(
    const float* __restrict__ Abuf, const float* __restrict__ Xin,
    const float* __restrict__ Wt, const float* __restrict__ bias,
    float* __restrict__ Out, int cin, int cout, int KcPad, int KPT,
    int nPairs) {
  int wave = threadIdx.x >> 5;
  int lane = threadIdx.x & 31;
  int np = blockIdx.y * 4 + wave;            // n-pair handled by this wave
  if (np >= nPairs) return;
  int n0 = np * 32;
  int m0 = blockIdx.x * 32;                  // m-pair base (2 tiles of 16)
  int half = lane >> 4;
  int mr = lane & 15;

  const float* a0row = Abuf + (size_t)(m0 + mr) * KcPad + 2 * half;
  const float* a1row = a0row + (size_t)16 * KcPad;
  const float* b0row = Wt + (size_t)(n0 + mr) * KPT + 2 * half;
  const float* b1row = b0row + (size_t)16 * KPT;

  v8f acc00 = {0.f, 0.f, 0.f, 0.f, 0.f, 0.f, 0.f, 0.f};
  v8f acc01 = {0.f, 0.f, 0.f, 0.f, 0.f, 0.f, 0.f, 0.f};
  v8f acc10 = {0.f, 0.f, 0.f, 0.f, 0.f, 0.f, 0.f, 0.f};
  v8f acc11 = {0.f, 0.f, 0.f, 0.f, 0.f, 0.f, 0.f, 0.f};

  for (int k = 0; k < KcPad; k += 4) {
    v2f a0 = *(const v2f*)(a0row + k);
    v2f a1 = *(const v2f*)(a1row + k);
    v2f b0 = *(const v2f*)(b0row + k);
    v2f b1 = *(const v2f*)(b1row + k);
    acc00 = __builtin_amdgcn_wmma_f32_16x16x4_f32(false, a0, false, b0,
                                                  (short)0, acc00, false, false);
    acc01 = __builtin_amdgcn_wmma_f32_16x16x4_f32(false, a0, false, b1,
                                                  (short)0, acc01, false, false);
    acc10 = __builtin_amdgcn_wmma_f32_16x16x4_f32(false, a1, false, b0,
                                                  (short)0, acc10, false, false);
    acc11 = __builtin_amdgcn_wmma_f32_16x16x4_f32(false, a1, false, b1,
                                                  (short)0, acc11, false, false);
  }

  // root-term: augment K with cin columns of X against Wt[:, KcPad:]
  int row0 = m0 + mr;
  int row1 = m0 + 16 + mr;
  const float* x0 = Xin + (size_t)row0 * cin;
  const float* x1 = Xin + (size_t)row1 * cin;
  bool r1ok = (row1 < N_NODES);
  int CinPad = KPT - KcPad;
  for (int c = 0; c < CinPad; c += 4) {
    int cc = c + 2 * half;
    bool c0ok = (cc < cin), c1ok = (cc + 1 < cin);
    v2f a0, a1;
    a0.x = c0ok ? x0[cc] : 0.0f;
    a0.y = c1ok ? x0[cc + 1] : 0.0f;
    a1.x = (r1ok && c0ok) ? x1[cc] : 0.0f;
    a1.y = (r1ok && c1ok) ? x1[cc + 1] : 0.0f;
    v2f b0 = *(const v2f*)(b0row + KcPad + c);
    v2f b1 = *(const v2f*)(b1row + KcPad + c);
    acc00 = __builtin_amdgcn_wmma_f32_16x16x4_f32(false, a0, false, b0,
                                                  (short)0, acc00, false, false);
    acc01 = __builtin_amdgcn_wmma_f32_16x16x4_f32(false, a0, false, b1,
                                                  (short)0, acc01, false, false);
    acc10 = __builtin_amdgcn_wmma_f32_16x16x4_f32(false, a1, false, b0,
                                                  (short)0, acc10, false, false);
    acc11 = __builtin_amdgcn_wmma_f32_16x16x4_f32(false, a1, false, b1,
                                                  (short)0, acc11, false, false);
  }

  int col0 = n0 + mr, col1 = n0 + 16 + mr;
  float bs0 = (col0 < cout) ? bias[col0] : 0.0f;
  float bs1 = (col1 < cout) ? bias[col1] : 0.0f;
#pragma unroll
  for (int v = 0; v < 8; ++v) {
    int r0 = m0 + v + 8 * half;              // C/D layout: M = v + 8*half
    int r1 = r0 + 16;
    if (col0 < cout) Out[(size_t)r0 * cout + col0] = acc00[v] + bs0;
    if (col1 < cout) Out[(size_t)r0 * cout + col1] = acc01[v] + bs1;
    if (r1 < N_NODES) {
      if (col0 < cout) Out[(size_t)r1 * cout + col0] = acc10[v] + bs0;
      if (col1 < cout) Out[(size_t)r1 * cout + col1] = acc11[v] + bs1;
    }
  }
}

// ---------------------------------------------------------------------------
// BN statistics over elu(H): one block per channel.
// ---------------------------------------------------------------------------
__global__ void bn_stats_kernel(const float* __restrict__ H,
                                float* __restrict__ stats, int cout) {
  int ch = blockIdx.x;
  int tid = threadIdx.x;
  __shared__ float ss[256], sq[256];
  float s = 0.f, q = 0.f;
  for (int n = tid; n < N_NODES; n += 256) {
    float h = H[(size_t)n * cout + ch];
    h = (h > 0.f) ? h : (expf(h) - 1.0f);
    s += h;
    q += h * h;
  }
  ss[tid] = s;
  sq[tid] = q;
  __syncthreads();
  for (int st = 128; st > 0; st >>= 1) {
    if (tid < st) { ss[tid] += ss[tid + st]; sq[tid] += sq[tid + st]; }
    __syncthreads();
  }
  if (tid == 0) {
    float m = ss[0] * (1.0f / N_NODES);
    float var = sq[0] * (1.0f / N_NODES) - m * m;
    stats[ch] = m;
    stats[cout + ch] = var;
  }
}

__global__ void bn_apply_kernel(float* __restrict__ H,
                                const float* __restrict__ stats,
                                const float* __restrict__ gamma,
                                const float* __restrict__ beta,
                                int cout, int total) {
  int idx = blockIdx.x * blockDim.x + threadIdx.x;
  if (idx >= total) return;
  int ch = idx % cout;
  float h = H[idx];
  h = (h > 0.f) ? h : (expf(h) - 1.0f);      // ELU then BN
  float m = stats[ch], v = stats[cout + ch];
  H[idx] = gamma[ch] * (h - m) * rsqrtf(v + EPS_BN) + beta[ch];
}

// ---------------------------------------------------------------------------
static inline size_t roundUpSz(size_t v, size_t a) { return (v + a - 1) / a * a; }
static inline int roundUpI(int v, int a) { return (v + a - 1) / a * a; }

extern "C" void kernel_launch(void* const* d_in, const int* in_sizes, int n_in,
                              void* d_out, int out_size, void* d_ws, size_t ws_size,
                              hipStream_t stream) {
  static const int chans[13] = {50, 75, 100, 200, 200, 200,
                                200, 200, 200, 200, 100, 75, 50};
  const float* x     = (const float*)d_in[0];
  const int*   eidx  = (const int*)d_in[1];
  const float* eattr = (const float*)d_in[2];
  const float* convW[12];
  const float* convRoot[12];
  const float* convBias[12];
  for (int i = 0; i < 12; ++i) {
    convW[i]    = (const float*)d_in[3 + 3 * i];
    convRoot[i] = (const float*)d_in[4 + 3 * i];
    convBias[i] = (const float*)d_in[5 + 3 * i];
  }
  const float* bnG[11];
  const float* bnB[11];
  for (int i = 0; i < 11; ++i) {
    bnG[i] = (const float*)d_in[39 + 2 * i];
    bnB[i] = (const float*)d_in[40 + 2 * i];
  }

  // Workspace carve-out (deterministic, same every call).
  size_t off = 0;
  auto carve = [&](size_t bytes) -> void* {
    void* p = (char*)d_ws + off;
    off = roundUpSz(off + bytes, 256);
    return p;
  };
  float* eb_w  = (float*)carve((size_t)N_EDGES * 8 * sizeof(float));
  int*   eb_dk = (int*)carve((size_t)N_EDGES * 8 * sizeof(int));
  float* Wt    = (float*)carve((size_t)224 * 5600 * sizeof(float));
  float* Ha    = (float*)carve((size_t)N_NODES * 200 * sizeof(float));
  float* Hb    = (float*)carve((size_t)N_NODES * 200 * sizeof(float));
  float* stats = (float*)carve(512 * sizeof(float));
  float* Abuf  = (float*)carve((size_t)N_NODES_PAD * 5400 * sizeof(float));

  edge_basis_kernel<<<(N_EDGES + 255) / 256, 256, 0, stream>>>(eattr, eidx,
                                                               eb_w, eb_dk);

  const float* hin = x;
  for (int i = 0; i < 12; ++i) {
    int cin = chans[i], cout = chans[i + 1];
    int Kc = KBINS * cin;
    int KcPad = roundUpI(Kc, 4);
    int CinPad = roundUpI(cin, 4);
    int KPT = KcPad + CinPad;
    int Npad = roundUpI(cout, 32);
    int nPairs = Npad / 32;

    // zero padded A buffer (incl. rows 10000..10015 used by last m-pair)
    hipMemsetAsync(Abuf, 0, (size_t)N_NODES_PAD * KcPad * sizeof(float),
                   stream);
    scatter_kernel<<<N_EDGES * 8, 64, 0, stream>>>(hin, eb_w, eb_dk, eidx,
                                                   Abuf, cin, KcPad);
    int wtTotal = Npad * KPT;
    build_wt_kernel<<<(wtTotal + 255) / 256, 256, 0, stream>>>(
        convW[i], convRoot[i], Wt, cout, cin, Kc, KcPad, KPT, wtTotal);

    float* outbuf = (i == 11) ? (float*)d_out : ((i % 2 == 0) ? Ha : Hb);
    dim3 grid(N_NODES_PAD / 32, (nPairs + 3) / 4);
    spline_gemm_wmma<<<grid, 128, 0, stream>>>(Abuf, hin, Wt, convBias[i],
                                               outbuf, cin, cout, KcPad, KPT,
                                               nPairs);

    if (i < 11) {
      bn_stats_kernel<<<cout, 256, 0, stream>>>(outbuf, stats, cout);
      int tot = N_NODES * cout;
      bn_apply_kernel<<<(tot + 255) / 256, 256, 0, stream>>>(
          outbuf, stats, bnG[i], bnB[i], cout, tot);
      hin = outbuf;
    }
  }
  (void)in_sizes; (void)n_in; (void)out_size; (void)ws_size;
}